// ZcDC_2D_55113020342642
// MI455X (gfx1250) — compile-verified
//
#include <hip/hip_runtime.h>
#include <math.h>

// ============================================================================
// ZcDC 2D MRI CG data-consistency on MI455X (gfx1250).
// 2D centered FFTs as complex GEMMs on f32 WMMA (16x16x4), 32x32 output per
// wave (2x2 register blocking -> 0.75 loads/WMMA). Centered DFT matrix
// D[k][n] = exp(-2pi*i*(k-N/2)*(n-N/2)/N)/sqrt(N) is symmetric & unitary, so
// inverse = conj(D); conjugation handled by pre-negated imag planes.
// Deterministic (no atomics). Workspace: ~58 MB.
// ============================================================================

typedef float v2f __attribute__((ext_vector_type(2)));
typedef float v8f __attribute__((ext_vector_type(8)));

constexpr int Hh = 448, Ww = 224, Dd = 12, Cc = 10;
constexpr int HW = Hh * Ww;          // 100352
constexpr int NPIX = Dd * HW;        // 1204224
constexpr int RED_BLOCKS = 1024;

// ---------------------------------------------------------------------------
// f32 WMMA 16x16x4: D = A(16x4) * B(4x16) + C
// ---------------------------------------------------------------------------
static __device__ inline v8f wmma4(v2f a, v2f b, v8f c) {
  return __builtin_amdgcn_wmma_f32_16x16x4_f32(
      /*neg_a=*/false, a, /*neg_b=*/false, b,
      /*c_mod=*/(short)0, c, /*reuse_a=*/false, /*reuse_b=*/false);
}

// 32x32 complex accumulator: [mi][ni] 16x16 tiles, real+imag planes.
struct CTile32 {
  v8f r[2][2];
  v8f i[2][2];
};

// ---------------------------------------------------------------------------
// 32x32 complex GEMM block: C = A(MxK) * B(KxN), rows [rowM,rowM+32),
// cols [colN,colN+32). Conjugation = pass a pre-negated imaginary plane.
// Fragment layouts per CDNA5 ISA 7.12.2 (f32):
//   A 16x4 : lane m=lane%16; VGPR0 = K(2*half), VGPR1 = K(2*half+1)
//   B 4x16 : lane n=lane%16; VGPR0 = row K(2*half), VGPR1 = row K(2*half+1)
//   C/D    : VGPR j -> element [M = j + 8*half][N = lane%16]
// ---------------------------------------------------------------------------
static __device__ inline void cgemm32(
    const float* __restrict__ Ar, const float* __restrict__ Ai, int lda,
    const float* __restrict__ Br, const float* __restrict__ Bi, int ldb,
    int rowM, int colN, int K, CTile32& t)
{
  const int lane = threadIdx.x & 31;
  const int half = lane >> 4;
  const int lm   = lane & 15;
  const float* arp[2], * aip[2];
  arp[0] = Ar + (rowM + lm) * lda + 2 * half;       arp[1] = arp[0] + 16 * lda;
  aip[0] = Ai + (rowM + lm) * lda + 2 * half;       aip[1] = aip[0] + 16 * lda;
  const float* brp = Br + (2 * half) * ldb + colN + lm;
  const float* bip = Bi + (2 * half) * ldb + colN + lm;

#pragma unroll 2
  for (int k0 = 0; k0 < K; k0 += 4) {
    v2f ar[2], ai[2], nai[2];
#pragma unroll
    for (int mi = 0; mi < 2; ++mi) {
      ar[mi]  = *(const v2f*)(arp[mi] + k0);
      ai[mi]  = *(const v2f*)(aip[mi] + k0);
      nai[mi] = -ai[mi];
    }
    const int r0 = k0 * ldb, r1 = (k0 + 1) * ldb;
    v2f br[2], bi[2];
#pragma unroll
    for (int ni = 0; ni < 2; ++ni) {
      br[ni].x = brp[r0 + 16 * ni];  br[ni].y = brp[r1 + 16 * ni];
      bi[ni].x = bip[r0 + 16 * ni];  bi[ni].y = bip[r1 + 16 * ni];
    }
    // (Ar + iAi)(Br + iBi) = (ArBr - AiBi) + i(ArBi + AiBr)
#pragma unroll
    for (int mi = 0; mi < 2; ++mi)
#pragma unroll
      for (int ni = 0; ni < 2; ++ni) {
        t.r[mi][ni] = wmma4(ar[mi],  br[ni], t.r[mi][ni]);
        t.r[mi][ni] = wmma4(nai[mi], bi[ni], t.r[mi][ni]);
        t.i[mi][ni] = wmma4(ar[mi],  bi[ni], t.i[mi][ni]);
        t.i[mi][ni] = wmma4(ai[mi],  br[ni], t.i[mi][ni]);
      }
  }
}

static __device__ inline void ctile_zero(CTile32& t) {
#pragma unroll
  for (int mi = 0; mi < 2; ++mi)
#pragma unroll
    for (int ni = 0; ni < 2; ++ni) {
      t.r[mi][ni] = (v8f){};
      t.i[mi][ni] = (v8f){};
    }
}

// ---------------------------------------------------------------------------
// Centered DFT matrix (+ pre-negated imaginary plane for the inverse/conj).
// Integer phase reduction keeps full fp32 accuracy.
// ---------------------------------------------------------------------------
__global__ void k_dftmat(int n, float* __restrict__ dr, float* __restrict__ di,
                         float* __restrict__ din) {
  int idx = blockIdx.x * blockDim.x + threadIdx.x;
  if (idx >= n * n) return;
  int k = idx / n, m = idx % n;
  int t = ((k - n / 2) * (m - n / 2)) % n;
  if (t < 0) t += n;
  float th = (float)t * (6.28318530717958647692f / (float)n);
  float s, c;
  sincosf(th, &s, &c);
  float rs = rsqrtf((float)n);
  dr[idx]  = c * rs;
  di[idx]  = -s * rs;
  din[idx] = s * rs;
}

__global__ void k_scal0(const float* __restrict__ miu, float* __restrict__ S) {
  if (threadIdx.x == 0 && blockIdx.x == 0) S[0] = fabsf(miu[0]);
}

// p = zf_c + mu*z_c ; r = p ; b = 0
__global__ void k_init(const float* __restrict__ z, const float* __restrict__ zf,
                       const float* __restrict__ S,
                       float* __restrict__ Pr, float* __restrict__ Pi,
                       float* __restrict__ Rr, float* __restrict__ Ri,
                       float* __restrict__ Br, float* __restrict__ Bi) {
  int i = blockIdx.x * blockDim.x + threadIdx.x;
  if (i >= NPIX) return;
  float mu = S[0];
  float pr = zf[i] + mu * z[i];
  float pi = zf[NPIX + i] + mu * z[NPIX + i];
  Pr[i] = pr; Pi[i] = pi;
  Rr[i] = pr; Ri[i] = pi;
  Br[i] = 0.0f; Bi[i] = 0.0f;
}

// t1[c] = coil[c] * p_d  (elementwise complex)
__global__ void k_cexpand(const float* __restrict__ cr, const float* __restrict__ ci,
                          const float* __restrict__ pr, const float* __restrict__ pi,
                          float* __restrict__ t1r, float* __restrict__ t1i) {
  int i = blockIdx.x * blockDim.x + threadIdx.x;
  int c = blockIdx.y;
  float pv = pr[i], qv = pi[i];
  float a = cr[c * HW + i], b = ci[c * HW + i];
  t1r[c * HW + i] = a * pv - b * qv;
  t1i[c * HW + i] = a * qv + b * pv;
}

// out[c] = D(HxH) @ in[c](HxW); conj via Di pointer. 2 waves/block on M.
__global__ void __launch_bounds__(64)
k_dft_rows(const float* __restrict__ Dr, const float* __restrict__ Di,
           const float* __restrict__ Xr, const float* __restrict__ Xi,
           float* __restrict__ Or, float* __restrict__ Oi) {
  int wv = threadIdx.x >> 5;
  int rowM = (blockIdx.x * 2 + wv) * 32;
  int colN = blockIdx.y * 32;
  int c = blockIdx.z;
  CTile32 t; ctile_zero(t);
  cgemm32(Dr, Di, Hh, Xr + c * HW, Xi + c * HW, Ww, rowM, colN, Hh, t);
  int lane = threadIdx.x & 31, half = lane >> 4, lm = lane & 15;
  float* orp = Or + c * HW;
  float* oip = Oi + c * HW;
#pragma unroll
  for (int mi = 0; mi < 2; ++mi)
#pragma unroll
    for (int ni = 0; ni < 2; ++ni)
#pragma unroll
      for (int j = 0; j < 8; ++j) {
        int r = rowM + mi * 16 + 8 * half + j, cl = colN + ni * 16 + lm;
        orp[r * Ww + cl] = t.r[mi][ni][j];
        oip[r * Ww + cl] = t.i[mi][ni][j];
      }
}

// out[c] = mask .* (in[c](HxW) @ Dw(WxW))  (forward W-axis DFT + k-space mask)
__global__ void __launch_bounds__(64)
k_dft_cols_mask(const float* __restrict__ Xr, const float* __restrict__ Xi,
                const float* __restrict__ Dr, const float* __restrict__ Di,
                const int* __restrict__ mask,
                float* __restrict__ Or, float* __restrict__ Oi) {
  int wv = threadIdx.x >> 5;
  int rowM = (blockIdx.x * 2 + wv) * 32;
  int colN = blockIdx.y * 32;
  int c = blockIdx.z;
  CTile32 t; ctile_zero(t);
  cgemm32(Xr + c * HW, Xi + c * HW, Ww, Dr, Di, Ww, rowM, colN, Ww, t);
  int lane = threadIdx.x & 31, half = lane >> 4, lm = lane & 15;
  float* orp = Or + c * HW;
  float* oip = Oi + c * HW;
#pragma unroll
  for (int mi = 0; mi < 2; ++mi)
#pragma unroll
    for (int ni = 0; ni < 2; ++ni)
#pragma unroll
      for (int j = 0; j < 8; ++j) {
        int r = rowM + mi * 16 + 8 * half + j, cl = colN + ni * 16 + lm;
        float m = (float)mask[r * Ww + cl];
        orp[r * Ww + cl] = t.r[mi][ni][j] * m;
        oip[r * Ww + cl] = t.i[mi][ni][j] * m;
      }
}

// out[c] = conj(coil_c) .* (in[c](HxW) @ conj(Dw))   (pass DwI_neg as Di)
__global__ void __launch_bounds__(64)
k_coil_combine(const float* __restrict__ Xr, const float* __restrict__ Xi,
               const float* __restrict__ Dr, const float* __restrict__ Di,
               const float* __restrict__ cr, const float* __restrict__ ci,
               float* __restrict__ Or, float* __restrict__ Oi) {
  int wv = threadIdx.x >> 5;
  int rowM = (blockIdx.x * 2 + wv) * 32;
  int colN = blockIdx.y * 32;
  int c = blockIdx.z;
  CTile32 t; ctile_zero(t);
  cgemm32(Xr + c * HW, Xi + c * HW, Ww, Dr, Di, Ww, rowM, colN, Ww, t);
  int lane = threadIdx.x & 31, half = lane >> 4, lm = lane & 15;
#pragma unroll
  for (int mi = 0; mi < 2; ++mi)
#pragma unroll
    for (int ni = 0; ni < 2; ++ni)
#pragma unroll
      for (int j = 0; j < 8; ++j) {
        int r = rowM + mi * 16 + 8 * half + j, cl = colN + ni * 16 + lm;
        int idx = c * HW + r * Ww + cl;
        float a = cr[idx], b = ci[idx];
        float x = t.r[mi][ni][j], y = t.i[mi][ni][j];
        // conj(a+ib)*(x+iy) = (a*x + b*y) + i(a*y - b*x)
        Or[idx] = a * x + b * y;
        Oi[idx] = a * y - b * x;
      }
}

// q_d = mu*p_d + sum_c t1[c]   (deterministic coil reduction)
__global__ void k_qsum(const float* __restrict__ t1r, const float* __restrict__ t1i,
                       const float* __restrict__ pr, const float* __restrict__ pi,
                       const float* __restrict__ S,
                       float* __restrict__ qr, float* __restrict__ qi) {
  int i = blockIdx.x * blockDim.x + threadIdx.x;
  float sr = 0.f, si = 0.f;
#pragma unroll
  for (int c = 0; c < Cc; ++c) {
    sr += t1r[c * HW + i];
    si += t1i[c * HW + i];
  }
  float mu = S[0];
  qr[i] = sr + mu * pr[i];
  qi[i] = si + mu * pi[i];
}

// ---------------------------------------------------------------------------
// Deterministic two-stage reductions for CG scalars (no atomics).
// ---------------------------------------------------------------------------
static __device__ inline float blockReduce256(float v) {
  __shared__ float sm[256];
  __syncthreads();
  sm[threadIdx.x] = v;
  __syncthreads();
#pragma unroll
  for (int s = 128; s > 0; s >>= 1) {
    if ((int)threadIdx.x < s) sm[threadIdx.x] += sm[threadIdx.x + s];
    __syncthreads();
  }
  return sm[0];
}

// part[3b] = sum |r|^2 ; part[3b+1..2] = Re/Im sum q*conj(p)
__global__ void k_reduce_rq(const float* __restrict__ Rr, const float* __restrict__ Ri,
                            const float* __restrict__ Qr, const float* __restrict__ Qi,
                            const float* __restrict__ Pr, const float* __restrict__ Pi,
                            float* __restrict__ part) {
  float rr = 0.f, qre = 0.f, qim = 0.f;
  for (int i = blockIdx.x * blockDim.x + threadIdx.x; i < NPIX; i += gridDim.x * blockDim.x) {
    float rv = Rr[i], iv = Ri[i];
    rr += rv * rv + iv * iv;
    float a = Qr[i], b = Qi[i], x = Pr[i], y = Pi[i];
    qre += a * x + b * y;
    qim += b * x - a * y;
  }
  rr = blockReduce256(rr);
  qre = blockReduce256(qre);
  qim = blockReduce256(qim);
  if (threadIdx.x == 0) {
    part[3 * blockIdx.x] = rr;
    part[3 * blockIdx.x + 1] = qre;
    part[3 * blockIdx.x + 2] = qim;
  }
}

// S[1]=rr ; S[2..3]=alpha = rr*conj(qp)/|qp|^2
__global__ void k_alpha(const float* __restrict__ part, float* __restrict__ S) {
  float rr = 0.f, qre = 0.f, qim = 0.f;
  for (int i = threadIdx.x; i < RED_BLOCKS; i += 256) {
    rr += part[3 * i]; qre += part[3 * i + 1]; qim += part[3 * i + 2];
  }
  rr = blockReduce256(rr);
  qre = blockReduce256(qre);
  qim = blockReduce256(qim);
  if (threadIdx.x == 0) {
    float den = qre * qre + qim * qim;
    S[1] = rr;
    S[2] = rr * qre / den;
    S[3] = -rr * qim / den;
  }
}

// b += alpha*p ; r -= alpha*q ; part[b] = partial sum |r_new|^2
__global__ void k_update_br(const float* __restrict__ S,
                            const float* __restrict__ Pr, const float* __restrict__ Pi,
                            const float* __restrict__ Qr, const float* __restrict__ Qi,
                            float* __restrict__ Br, float* __restrict__ Bi,
                            float* __restrict__ Rr, float* __restrict__ Ri,
                            float* __restrict__ part) {
  float ar = S[2], ai = S[3];
  float acc = 0.f;
  for (int i = blockIdx.x * blockDim.x + threadIdx.x; i < NPIX; i += gridDim.x * blockDim.x) {
    float x = Pr[i], y = Pi[i];
    Br[i] += ar * x - ai * y;
    Bi[i] += ar * y + ai * x;
    float a = Qr[i], b = Qi[i];
    float nr = Rr[i] - (ar * a - ai * b);
    float ni = Ri[i] - (ar * b + ai * a);
    Rr[i] = nr; Ri[i] = ni;
    acc += nr * nr + ni * ni;
  }
  acc = blockReduce256(acc);
  if (threadIdx.x == 0) part[blockIdx.x] = acc;
}

__global__ void k_beta(const float* __restrict__ part, float* __restrict__ S) {
  float v = 0.f;
  for (int i = threadIdx.x; i < RED_BLOCKS; i += 256) v += part[i];
  v = blockReduce256(v);
  if (threadIdx.x == 0) S[4] = v / S[1];
}

// p = r + beta*p
__global__ void k_pupdate(const float* __restrict__ S,
                          const float* __restrict__ Rr, const float* __restrict__ Ri,
                          float* __restrict__ Pr, float* __restrict__ Pi) {
  int i = blockIdx.x * blockDim.x + threadIdx.x;
  if (i >= NPIX) return;
  float beta = S[4];
  Pr[i] = Rr[i] + beta * Pr[i];
  Pi[i] = Ri[i] + beta * Pi[i];
}

__global__ void k_output(const float* __restrict__ Br, const float* __restrict__ Bi,
                         float* __restrict__ out) {
  int i = blockIdx.x * blockDim.x + threadIdx.x;
  if (i >= NPIX) return;
  out[i] = Br[i];
  out[NPIX + i] = Bi[i];
}

// ===========================================================================
extern "C" void kernel_launch(void* const* d_in, const int* in_sizes, int n_in,
                              void* d_out, int out_size, void* d_ws, size_t ws_size,
                              hipStream_t stream) {
  const float* z   = (const float*)d_in[0];   // [1,2,D,H,W]
  const float* zf  = (const float*)d_in[1];   // [1,2,D,H,W]
  const float* cr  = (const float*)d_in[2];   // [1,C,1,H,W]
  const float* ci  = (const float*)d_in[3];   // [1,C,1,H,W]
  const int*   mk  = (const int*)d_in[4];     // [1,1,1,H,W]
  const float* miu = (const float*)d_in[5];   // [1]
  float* out = (float*)d_out;
  float* ws  = (float*)d_ws;

  size_t o = 0;
  float* S    = ws + o; o += 16;
  float* DhR  = ws + o; o += (size_t)Hh * Hh;
  float* DhI  = ws + o; o += (size_t)Hh * Hh;
  float* DhIn = ws + o; o += (size_t)Hh * Hh;   // -imag (conj/inverse)
  float* DwR  = ws + o; o += (size_t)Ww * Ww;
  float* DwI  = ws + o; o += (size_t)Ww * Ww;
  float* DwIn = ws + o; o += (size_t)Ww * Ww;   // -imag (conj/inverse)
  float* Pr = ws + o; o += NPIX;  float* Pi = ws + o; o += NPIX;
  float* Rr = ws + o; o += NPIX;  float* Ri = ws + o; o += NPIX;
  float* Br = ws + o; o += NPIX;  float* Bi = ws + o; o += NPIX;
  float* Qr = ws + o; o += NPIX;  float* Qi = ws + o; o += NPIX;
  float* T1r = ws + o; o += (size_t)Cc * HW;  float* T1i = ws + o; o += (size_t)Cc * HW;
  float* T2r = ws + o; o += (size_t)Cc * HW;  float* T2i = ws + o; o += (size_t)Cc * HW;
  float* Part = ws + o; o += 3 * RED_BLOCKS;
  // total ~14.5M floats (~58 MB)

  const dim3 gemmGrid(Hh / 64, Ww / 32, Cc);   // (7,7,C), 2 waves/block on M
  const int EPB = (NPIX + 255) / 256;

  k_scal0<<<1, 64, 0, stream>>>(miu, S);
  k_dftmat<<<(Hh * Hh + 255) / 256, 256, 0, stream>>>(Hh, DhR, DhI, DhIn);
  k_dftmat<<<(Ww * Ww + 255) / 256, 256, 0, stream>>>(Ww, DwR, DwI, DwIn);
  k_init<<<EPB, 256, 0, stream>>>(z, zf, S, Pr, Pi, Rr, Ri, Br, Bi);

  for (int it = 0; it < 5; ++it) {
    for (int d = 0; d < Dd; ++d) {
      const float* pdr = Pr + (size_t)d * HW;
      const float* pdi = Pi + (size_t)d * HW;
      float* qdr = Qr + (size_t)d * HW;
      float* qdi = Qi + (size_t)d * HW;
      // t1 = coil .* p_d
      k_cexpand<<<dim3(HW / 256, Cc), 256, 0, stream>>>(cr, ci, pdr, pdi, T1r, T1i);
      // t2 = Dh @ t1                     (forward H-axis centered DFT)
      k_dft_rows<<<gemmGrid, 64, 0, stream>>>(DhR, DhI, T1r, T1i, T2r, T2i);
      // t1 = mask .* (t2 @ Dw)           (forward W-axis DFT + sampling mask)
      k_dft_cols_mask<<<gemmGrid, 64, 0, stream>>>(T2r, T2i, DwR, DwI, mk, T1r, T1i);
      // t2 = conj(Dh) @ t1               (inverse H-axis)
      k_dft_rows<<<gemmGrid, 64, 0, stream>>>(DhR, DhIn, T1r, T1i, T2r, T2i);
      // t1 = conj(coil) .* (t2 @ conj(Dw))   (inverse W-axis + coil adjoint)
      k_coil_combine<<<gemmGrid, 64, 0, stream>>>(T2r, T2i, DwR, DwIn, cr, ci, T1r, T1i);
      // q_d = mu*p_d + sum_c t1[c]
      k_qsum<<<HW / 256, 256, 0, stream>>>(T1r, T1i, pdr, pdi, S, qdr, qdi);
    }
    k_reduce_rq<<<RED_BLOCKS, 256, 0, stream>>>(Rr, Ri, Qr, Qi, Pr, Pi, Part);
    k_alpha<<<1, 256, 0, stream>>>(Part, S);
    k_update_br<<<RED_BLOCKS, 256, 0, stream>>>(S, Pr, Pi, Qr, Qi, Br, Bi, Rr, Ri, Part);
    k_beta<<<1, 256, 0, stream>>>(Part, S);
    k_pupdate<<<EPB, 256, 0, stream>>>(S, Rr, Ri, Pr, Pi);
  }
  k_output<<<EPB, 256, 0, stream>>>(Br, Bi, out);
  (void)in_sizes; (void)n_in; (void)out_size; (void)ws_size;
}